// SimpleGPTModelKVCache_39247411150864
// MI455X (gfx1250) — compile-verified
//
#include <hip/hip_runtime.h>
#include <hip/hip_bf16.h>
#include <math.h>

// ---------------------------------------------------------------------------
// MI455X (gfx1250) GPT decode step, B=32, EMB=1024, CTX=1024, HID=4096,
// VOCAB=65535.  Memory-bound: ~580 MB of fp32 weights + KV cache per call
// => ~25us floor at 23.3 TB/s.  GEMMs use V_WMMA_F32_16X16X4_F32 (wave32,
// M=32 -> two 16x16 accumulator tiles per wave).  Inner loop is kept free of
// EXEC manipulation: B columns are clamped (always-valid loads), stores are
// guarded only in the epilogue, and the W pointer is strength-reduced.
// ---------------------------------------------------------------------------

typedef __attribute__((ext_vector_type(2))) float v2f;
typedef __attribute__((ext_vector_type(8))) float v8f;

#define EMB   1024
#define CTXm1 1023
#define BATCH 32
#define HID   4096
#define VOCAB 65535
#define KCHUNK 64
#define LDSSTR (KCHUNK + 1)   // padded LDS stride: avoids 16-way bank conflicts

__device__ __forceinline__ float gelu_exact(float x) {
    return 0.5f * x * (1.0f + erff(x * 0.70710678118654752f));
}

// ----------------------------- embedding -----------------------------------
__global__ void embed_kernel(const int* __restrict__ idx,
                             const float* __restrict__ emb,
                             const float* __restrict__ pos,
                             float* __restrict__ x) {
    int b = blockIdx.y;
    int e = blockIdx.x * 256 + threadIdx.x;
    int tok = idx[b];
    x[b * EMB + e] = emb[(long)tok * EMB + e] + pos[e];
}

// ----------------- generic WMMA GEMM: C[32xN] = A[32xK] @ W[KxN] + bias ----
// 4 waves / block, each wave owns a 32x16 output tile (2 wmma accumulators
// sharing one B fragment).  A is staged through LDS (coalesced, padded).
// W is streamed straight from HBM (the bandwidth-critical operand) through a
// running per-lane pointer: no per-iteration 64-bit multiplies, no EXEC
// save/restore in the hot loop.
template <int ACT>  // 0 = none, 1 = exact gelu
__global__ __launch_bounds__(128)
void wmma_gemm_m32(const float* __restrict__ A,
                   const float* __restrict__ W,
                   const float* __restrict__ bias,
                   float* __restrict__ C,
                   int K, int N) {
    __shared__ float As[32 * LDSSTR];

    const int tid  = threadIdx.x;
    const int lane = tid & 31;
    const int wave = tid >> 5;                    // 0..3
    const int n0   = blockIdx.x * 64 + wave * 16; // 16 cols per wave
    const int half = lane >> 4;                   // 0: K pair {k,k+1}, 1: {k+2,k+3}
    const int l16  = lane & 15;
    const int  col   = n0 + l16;
    const bool colOK = (col < N);
    // Clamped column: every W load is in-bounds and unconditional; lanes with
    // col >= N compute garbage that is simply never stored.
    const int colc = colOK ? col : (N - 1);
    const size_t sN = (size_t)N;

    // per-lane running pointer into W: row (half?2:0), column colc
    const float* pw = W + (half ? 2 * sN : 0) + (size_t)colc;

    v8f acc0 = {};   // rows 0..15
    v8f acc1 = {};   // rows 16..31

    for (int kc = 0; kc < K; kc += KCHUNK) {
        __syncthreads();
        // cooperative, coalesced copy of A[32][KCHUNK] into LDS
        for (int i = tid; i < 32 * KCHUNK; i += 128) {
            int r  = i / KCHUNK;
            int kk = i - r * KCHUNK;
            As[r * LDSSTR + kk] = A[(long)r * K + kc + kk];
        }
        __syncthreads();

        #pragma unroll 4
        for (int k = 0; k < KCHUNK; k += 4) {
            const int ka = k + (half ? 2 : 0);
            v2f a0, a1, b;
            // A fragment (16x4): lanes 0-15 hold K={k,k+1}, lanes 16-31 K={k+2,k+3}
            a0.x = As[l16 * LDSSTR + ka];
            a0.y = As[l16 * LDSSTR + ka + 1];
            a1.x = As[(l16 + 16) * LDSSTR + ka];
            a1.y = As[(l16 + 16) * LDSSTR + ka + 1];
            // B fragment (4x16): row striped across lanes within a VGPR
            b.x = pw[0];
            b.y = pw[sN];
            pw += 4 * sN;

            acc0 = __builtin_amdgcn_wmma_f32_16x16x4_f32(
                false, a0, false, b, (short)0, acc0, false, false);
            acc1 = __builtin_amdgcn_wmma_f32_16x16x4_f32(
                false, a1, false, b, (short)0, acc1, false, false);
        }
    }

    if (colOK) {
        const float bc = bias[col];
        #pragma unroll
        for (int r = 0; r < 8; ++r) {
            int m0 = r + (half ? 8 : 0);        // C/D layout: lanes 16-31 = M+8
            float x0 = acc0[r] + bc;
            float x1 = acc1[r] + bc;
            if (ACT == 1) { x0 = gelu_exact(x0); x1 = gelu_exact(x1); }
            C[(long)m0 * N + col]        = x0;
            C[(long)(m0 + 16) * N + col] = x1;
        }
    }
}

// --------------------- attention: scores = q . K^T * scale -----------------
// block = (batch, s-chunk of 8); wave-per-row, fully coalesced 512B reads.
__global__ __launch_bounds__(256)
void attn_scores_kernel(const float* __restrict__ q_new,
                        const float* __restrict__ key_cache,
                        const float* __restrict__ k_new,
                        float* __restrict__ scores) {
    __shared__ float qs[EMB];
    const int b   = blockIdx.x;
    const int tid = threadIdx.x;
    for (int i = tid; i < EMB; i += 256) qs[i] = q_new[b * EMB + i];
    __syncthreads();

    const int wave = tid >> 5, lane = tid & 31;
    const int s = blockIdx.y * 8 + wave;
    const float* krow = (s < CTXm1)
        ? (key_cache + ((long)b * CTXm1 + s) * EMB)
        : (k_new + (long)b * EMB);

    float acc = 0.0f;
    #pragma unroll
    for (int i = 0; i < 8; ++i) {
        int d = i * 128 + lane * 4;
        float4 kv = *(const float4*)(krow + d);
        acc += kv.x * qs[d] + kv.y * qs[d + 1] + kv.z * qs[d + 2] + kv.w * qs[d + 3];
    }
    #pragma unroll
    for (int off = 16; off > 0; off >>= 1) acc += __shfl_xor(acc, off, 32);
    if (lane == 0) scores[b * 1024 + s] = acc * 0.03125f;  // 1/sqrt(1024)
}

// ------------------------------ softmax ------------------------------------
__global__ __launch_bounds__(256)
void softmax_kernel(float* __restrict__ scores) {
    __shared__ float redmax[8];
    __shared__ float redsum[8];
    const int b = blockIdx.x, tid = threadIdx.x;
    const int wave = tid >> 5, lane = tid & 31;
    float* row = scores + b * 1024;

    float v[4];
    float m = -1e30f;
    #pragma unroll
    for (int i = 0; i < 4; ++i) { v[i] = row[tid + 256 * i]; m = fmaxf(m, v[i]); }
    #pragma unroll
    for (int off = 16; off > 0; off >>= 1) m = fmaxf(m, __shfl_xor(m, off, 32));
    if (lane == 0) redmax[wave] = m;
    __syncthreads();
    float gm = redmax[0];
    #pragma unroll
    for (int i = 1; i < 8; ++i) gm = fmaxf(gm, redmax[i]);

    float s = 0.0f;
    #pragma unroll
    for (int i = 0; i < 4; ++i) { v[i] = __expf(v[i] - gm); s += v[i]; }
    #pragma unroll
    for (int off = 16; off > 0; off >>= 1) s += __shfl_xor(s, off, 32);
    if (lane == 0) redsum[wave] = s;
    __syncthreads();
    float gs = 0.0f;
    #pragma unroll
    for (int i = 0; i < 8; ++i) gs += redsum[i];
    const float inv = 1.0f / gs;
    #pragma unroll
    for (int i = 0; i < 4; ++i) row[tid + 256 * i] = v[i] * inv;
}

// ----------------- attn_out partials: probs @ V (s-chunked) ----------------
// Uniform-per-iteration branch is hoisted: cache rows first, then (only for
// the final chunk) the single freshly-computed v_new row.
__global__ __launch_bounds__(256)
void attn_av_partial(const float* __restrict__ probs,
                     const float* __restrict__ value_cache,
                     const float* __restrict__ v_new,
                     float* __restrict__ part) {
    __shared__ float ps[128];
    const int b = blockIdx.x, chunk = blockIdx.y, tid = threadIdx.x;
    const int s0 = chunk * 128;
    if (tid < 128) ps[tid] = probs[b * 1024 + s0 + tid];
    __syncthreads();

    const int d = tid * 4;
    float4 acc = make_float4(0.f, 0.f, 0.f, 0.f);

    const int nCache = (s0 + 128 <= CTXm1) ? 128 : (CTXm1 - s0);  // 128 or 127
    const float* vrow = value_cache + ((long)b * CTXm1 + s0) * EMB + d;
    for (int i = 0; i < nCache; ++i) {
        const float p = ps[i];
        float4 vv = *(const float4*)vrow;
        acc.x += p * vv.x; acc.y += p * vv.y; acc.z += p * vv.z; acc.w += p * vv.w;
        vrow += EMB;
    }
    if (nCache == 127) {  // last chunk: s = 1023 comes from v_new
        const float p = ps[127];
        float4 vv = *(const float4*)(v_new + (long)b * EMB + d);
        acc.x += p * vv.x; acc.y += p * vv.y; acc.z += p * vv.z; acc.w += p * vv.w;
    }
    *(float4*)(part + ((long)chunk * BATCH + b) * EMB + d) = acc;
}

__global__ void attn_av_reduce(const float* __restrict__ part,
                               float* __restrict__ out) {
    const int b = blockIdx.x;
    const int d = blockIdx.y * 256 + threadIdx.x;
    float acc = 0.0f;
    #pragma unroll
    for (int c = 0; c < 8; ++c) acc += part[((long)c * BATCH + b) * EMB + d];
    out[b * EMB + d] = acc;
}

// ---------------------------------------------------------------------------
extern "C" void kernel_launch(void* const* d_in, const int* in_sizes, int n_in,
                              void* d_out, int out_size, void* d_ws, size_t ws_size,
                              hipStream_t stream) {
    const int*   idx         = (const int*)  d_in[0];
    const float* key_cache   = (const float*)d_in[1];
    const float* value_cache = (const float*)d_in[2];
    const float* emb         = (const float*)d_in[3];
    const float* pos         = (const float*)d_in[4];
    const float* Wk = (const float*)d_in[5];  const float* bk = (const float*)d_in[6];
    const float* Wv = (const float*)d_in[7];  const float* bv = (const float*)d_in[8];
    const float* Wq = (const float*)d_in[9];  const float* bq = (const float*)d_in[10];
    const float* W1 = (const float*)d_in[11]; const float* b1 = (const float*)d_in[12];
    const float* W2 = (const float*)d_in[13]; const float* b2 = (const float*)d_in[14];
    const float* Wl = (const float*)d_in[15]; const float* bl = (const float*)d_in[16];
    float* out = (float*)d_out;

    // workspace layout (floats): ~2.5 MB total
    float* ws     = (float*)d_ws;
    float* x      = ws;            // 32*1024
    float* kn     = ws + 32768;    // 32*1024
    float* vn     = ws + 65536;    // 32*1024
    float* qn     = ws + 98304;    // 32*1024
    float* scores = ws + 131072;   // 32*1024
    float* part   = ws + 163840;   // 8*32*1024
    float* attn   = ws + 425984;   // 32*1024
    float* h      = ws + 458752;   // 32*4096
    float* ffn    = ws + 589824;   // 32*1024

    // 1) x = emb_table[idx] + pos_table[0]
    embed_kernel<<<dim3(EMB / 256, BATCH), 256, 0, stream>>>(idx, emb, pos, x);

    // 2) QKV projections (WMMA, M=32, N=1024, K=1024)
    wmma_gemm_m32<0><<<dim3(EMB / 64), 128, 0, stream>>>(x, Wk, bk, kn, EMB, EMB);
    wmma_gemm_m32<0><<<dim3(EMB / 64), 128, 0, stream>>>(x, Wv, bv, vn, EMB, EMB);
    wmma_gemm_m32<0><<<dim3(EMB / 64), 128, 0, stream>>>(x, Wq, bq, qn, EMB, EMB);

    // 3) attention (streams the 268 MB KV cache exactly once, coalesced)
    attn_scores_kernel<<<dim3(BATCH, 128), 256, 0, stream>>>(qn, key_cache, kn, scores);
    softmax_kernel<<<BATCH, 256, 0, stream>>>(scores);
    attn_av_partial<<<dim3(BATCH, 8), 256, 0, stream>>>(scores, value_cache, vn, part);
    attn_av_reduce<<<dim3(BATCH, EMB / 256), 256, 0, stream>>>(part, attn);

    // 4) FFN: gelu(attn@W1+b1)@W2+b2
    wmma_gemm_m32<1><<<dim3(HID / 64), 128, 0, stream>>>(attn, W1, b1, h, EMB, HID);
    wmma_gemm_m32<0><<<dim3(EMB / 64), 128, 0, stream>>>(h, W2, b2, ffn, HID, EMB);

    // 5) logits = ffn @ Wl + bl   (N=65535 tail handled by column clamp+guard)
    wmma_gemm_m32<0><<<dim3((VOCAB + 63) / 64), 128, 0, stream>>>(ffn, Wl, bl, out, EMB, VOCAB);
}